// Spatial_STMoE_Light_38182259261878
// MI455X (gfx1250) — compile-verified
//
#include <hip/hip_runtime.h>
#include <math.h>

typedef __attribute__((ext_vector_type(2))) float v2f;
typedef __attribute__((ext_vector_type(8))) float v8f;

#define DEVI __device__ __forceinline__

#define BB   4
#define NQL  2048
#define DD   256
#define D2   512
#define HID  1365
#define HPAD 1376
#define NEXP 16
#define CAPC 256
#define QCC  256

DEVI float gelu_f(float x){ return 0.5f*x*(1.0f + erff(x*0.70710678118654752440f)); }

template<int ACT> DEVI float act_f(float x){
  if (ACT==1) return gelu_f(x);
  if (ACT==2) return x > 0.f ? x : 0.01f*x;
  return x;
}

DEVI float wave_sum(float v){
  #pragma unroll
  for (int o=16;o>0;o>>=1) v += __shfl_xor(v,o,32);
  return v;
}
DEVI float wave_max(float v){
  #pragma unroll
  for (int o=16;o>0;o>>=1) v = fmaxf(v, __shfl_xor(v,o,32));
  return v;
}

// fp32 WMMA: D(16x16) = A(16x4) * B(4x16) + C   (wave32, per CDNA5 ISA layouts)
DEVI v8f wmma4(v2f a, v2f b, v8f c){
  return __builtin_amdgcn_wmma_f32_16x16x4_f32(false, a, false, b, (short)0, c, false, false);
}
DEVI v8f zero8(){ v8f z = {0.f,0.f,0.f,0.f,0.f,0.f,0.f,0.f}; return z; }

// ---------------- generic batched GEMM: C = act(A*B + bias) -------------------
// Wave computes a 32x32 C tile (2x2 WMMA fragments -> 4 WMMAs per fragment load pair).
// A: [M,K] lda ; B: [K,N] ldb ; per-z strides sA/sB/sC, bias stride sBias.
// grid: x = ceil(N/32), y = ceil(M/256), z = batch ; block 256 (8 waves)
template<int ACT>
__global__ __launch_bounds__(256) void gemm_wmma(
    const float* __restrict__ A, int lda, long sA,
    const float* __restrict__ Bm, int ldb, long sB,
    const float* __restrict__ bias, int sBias,
    float* __restrict__ C, int ldc, long sC,
    int M, int N, int K)
{
  int z = blockIdx.z;
  A  += (long)z*sA;  Bm += (long)z*sB;  C += (long)z*sC;
  int wave = threadIdx.x>>5, lane = threadIdx.x&31;
  int half = lane>>4, r = lane&15;
  int m0 = (blockIdx.y*8 + wave)*32;
  int n0 = blockIdx.x*32;
  if (m0 >= M) return;
  int am0 = m0 + r,      am1 = m0 + 16 + r;
  int bn0 = n0 + r,      bn1 = n0 + 16 + r;
  bool mok0 = am0 < M, mok1 = am1 < M;
  bool nok0 = bn0 < N, nok1 = bn1 < N;
  const float* Ar0 = A + (long)(mok0?am0:0)*lda;
  const float* Ar1 = A + (long)(mok1?am1:0)*lda;
  v8f acc00 = zero8(), acc01 = zero8(), acc10 = zero8(), acc11 = zero8();
  #pragma unroll 2
  for (int k=0;k<K;k+=4){
    int k0 = k + half*2;
    bool kk0 = k0 < K, kk1 = (k0+1) < K;
    v2f a0,a1,b0,b1;
    a0.x = (mok0 && kk0) ? Ar0[k0]   : 0.f;
    a0.y = (mok0 && kk1) ? Ar0[k0+1] : 0.f;
    a1.x = (mok1 && kk0) ? Ar1[k0]   : 0.f;
    a1.y = (mok1 && kk1) ? Ar1[k0+1] : 0.f;
    const float* B0 = Bm + (long)k0*ldb;
    b0.x = (nok0 && kk0) ? B0[bn0]       : 0.f;
    b0.y = (nok0 && kk1) ? B0[ldb + bn0] : 0.f;
    b1.x = (nok1 && kk0) ? B0[bn1]       : 0.f;
    b1.y = (nok1 && kk1) ? B0[ldb + bn1] : 0.f;
    acc00 = wmma4(a0,b0,acc00);
    acc01 = wmma4(a0,b1,acc01);
    acc10 = wmma4(a1,b0,acc10);
    acc11 = wmma4(a1,b1,acc11);
  }
  float bv0 = (bias != nullptr && nok0) ? bias[(long)z*sBias + bn0] : 0.f;
  float bv1 = (bias != nullptr && nok1) ? bias[(long)z*sBias + bn1] : 0.f;
  #pragma unroll
  for (int i=0;i<8;i++){
    int row0 = m0 + half*8 + i;
    int row1 = row0 + 16;
    if (row0 < M){
      if (nok0) C[(long)row0*ldc + bn0] = act_f<ACT>(acc00[i] + bv0);
      if (nok1) C[(long)row0*ldc + bn1] = act_f<ACT>(acc01[i] + bv1);
    }
    if (row1 < M){
      if (nok0) C[(long)row1*ldc + bn0] = act_f<ACT>(acc10[i] + bv0);
      if (nok1) C[(long)row1*ldc + bn1] = act_f<ACT>(acc11[i] + bv1);
    }
  }
}

// ---------------- attention -------------------
// S[z=b*8+h][q][n] = (QH . KH) / sqrt(32); wave does 32 queries x 32 keys
// grid: x = Nk/32, z = B*8 ; 8 waves cover QCC=256 queries
__global__ __launch_bounds__(256) void attn_scores(
   const float* __restrict__ QH, const float* __restrict__ KH,
   float* __restrict__ S, int NQtot, int Nk, int q0)
{
  int z = blockIdx.z; int b = z>>3, h = z&7;
  int wave = threadIdx.x>>5, lane=threadIdx.x&31, half=lane>>4, r=lane&15;
  int q_lo = q0 + wave*32;
  int n0 = blockIdx.x*32 + r, n1 = n0 + 16;
  const float* Aq0 = QH + ((long)(b*NQtot + q_lo + r))*DD + h*32;
  const float* Aq1 = Aq0 + 16L*DD;
  const float* Kn0 = KH + ((long)b*Nk + n0)*DD + h*32;
  const float* Kn1 = Kn0 + 16L*DD;
  v8f acc00 = zero8(), acc01 = zero8(), acc10 = zero8(), acc11 = zero8();
  #pragma unroll
  for(int k=0;k<32;k+=4){
    int k0=k+half*2;
    v2f a0,a1,b0,b1;
    a0.x = Aq0[k0]; a0.y = Aq0[k0+1];
    a1.x = Aq1[k0]; a1.y = Aq1[k0+1];
    b0.x = Kn0[k0]; b0.y = Kn0[k0+1];
    b1.x = Kn1[k0]; b1.y = Kn1[k0+1];
    acc00 = wmma4(a0,b0,acc00);
    acc01 = wmma4(a0,b1,acc01);
    acc10 = wmma4(a1,b0,acc10);
    acc11 = wmma4(a1,b1,acc11);
  }
  float* Sz = S + (long)z*QCC*Nk;
  const float sc = 0.17677669529663688f; // 1/sqrt(32)
  #pragma unroll
  for(int i=0;i<8;i++){
    int row0 = wave*32 + half*8 + i;
    int row1 = row0 + 16;
    Sz[(long)row0*Nk + n0] = acc00[i]*sc;
    Sz[(long)row0*Nk + n1] = acc01[i]*sc;
    Sz[(long)row1*Nk + n0] = acc10[i]*sc;
    Sz[(long)row1*Nk + n1] = acc11[i]*sc;
  }
}

__global__ __launch_bounds__(256) void softmax_rows(
  float* __restrict__ S, int Nk, const unsigned char* __restrict__ mask)
{
  int wave=threadIdx.x>>5, lane=threadIdx.x&31;
  int rid = blockIdx.x*8 + wave;     // rid < 32*QCC
  int z = rid / QCC; int b = z>>3;
  float* row = S + (long)rid*Nk;
  const unsigned char* mrow = mask ? (mask + (long)b*Nk) : nullptr;
  float mx = -3.0e38f;
  for(int j=lane;j<Nk;j+=32){
    float v = row[j];
    if (mrow && !mrow[j]) v = -1e30f;
    mx = fmaxf(mx, v);
  }
  mx = wave_max(mx);
  float sum=0.f;
  for(int j=lane;j<Nk;j+=32){
    float e = (mrow && !mrow[j]) ? 0.f : expf(row[j] - mx);
    row[j] = e; sum += e;
  }
  sum = wave_sum(sum);
  float inv = 1.0f / fmaxf(sum, 1e-30f);
  for(int j=lane;j<Nk;j+=32) row[j] *= inv;
}

// ATT[b,q0+q,h*32+n] = sum_k S[z][q][k] * VH[b,k,h*32+n]
// wave does 32 queries x 32 head-dims (N=32 exactly); grid: (1,1,B*8)
__global__ __launch_bounds__(256) void attn_av(
   const float* __restrict__ S, const float* __restrict__ VH,
   float* __restrict__ ATT, int NQtot, int Nk, int q0)
{
  int z=blockIdx.z; int b=z>>3, h=z&7;
  int wave=threadIdx.x>>5, lane=threadIdx.x&31, half=lane>>4, r=lane&15;
  const float* S0 = S + (long)z*QCC*Nk + (long)(wave*32 + r)*Nk;
  const float* S1 = S0 + 16L*Nk;
  const float* V0 = VH + ((long)b*Nk)*DD + h*32 + r;
  const float* V1 = V0 + 16;
  v8f acc00 = zero8(), acc01 = zero8(), acc10 = zero8(), acc11 = zero8();
  #pragma unroll 2
  for(int k=0;k<Nk;k+=4){
    int k0=k+half*2;
    v2f a0,a1,b0,b1;
    a0.x = S0[k0]; a0.y = S0[k0+1];
    a1.x = S1[k0]; a1.y = S1[k0+1];
    const float* Vk = V0 + (long)k0*DD;
    b0.x = Vk[0];   b0.y = Vk[DD];
    b1.x = Vk[16];  b1.y = Vk[DD+16];
    acc00 = wmma4(a0,b0,acc00);
    acc01 = wmma4(a0,b1,acc01);
    acc10 = wmma4(a1,b0,acc10);
    acc11 = wmma4(a1,b1,acc11);
  }
  #pragma unroll
  for(int i=0;i<8;i++){
    int row0 = wave*32 + half*8 + i;
    int row1 = row0 + 16;
    long base0 = ((long)(b*NQtot + q0 + row0))*DD + h*32 + r;
    long base1 = ((long)(b*NQtot + q0 + row1))*DD + h*32 + r;
    ATT[base0]      = acc00[i];
    ATT[base0 + 16] = acc01[i];
    ATT[base1]      = acc10[i];
    ATT[base1 + 16] = acc11[i];
  }
}

// ---------------- small-K / pointwise kernels -------------------
__global__ void topo_kernel(const float* __restrict__ src, int mode, int N, int stride,
                            const float* __restrict__ w, const float* __restrict__ bb,
                            float* __restrict__ out)
{
  long idx = (long)blockIdx.x*blockDim.x + threadIdx.x;
  long total = (long)BB*N*DD;
  if (idx>=total) return;
  int c = (int)(idx & 255); long rr = idx >> 8;
  int n = (int)(rr % N); int b = (int)(rr / N);
  float x0,x1,x2;
  if (mode==0){ const float* p = src + ((long)b*N + n)*stride; x0=p[0]; x1=p[1]; x2=p[2]; }
  else { const float* p = src + (long)b*64*2304 + n;
         x0=p[0]; x1=p[2304]; x2=p[2*2304]; }
  float v = x0*w[c] + x1*w[DD+c] + x2*w[2*DD+c] + bb[c];
  out[idx] = gelu_f(v);
}

__global__ void values_kernel(const float* __restrict__ src, int mode, int N, int Kin,
                              const float* __restrict__ w, const float* __restrict__ bb,
                              float* __restrict__ out)
{
  long idx = (long)blockIdx.x*blockDim.x + threadIdx.x;
  long total = (long)BB*N*DD;
  if (idx>=total) return;
  int c = (int)(idx & 255); long rr = idx >> 8;
  int n = (int)(rr % N); int b = (int)(rr / N);
  float acc = bb[c];
  if (mode==0){
    const float* p = src + ((long)b*N + n)*Kin;
    for (int j=0;j<Kin;j++) acc += p[j]*w[j*DD + c];
  } else {
    const float* p = src + (long)b*64*2304 + n;
    for (int j=0;j<Kin;j++) acc += p[(long)j*2304]*w[j*DD + c];
  }
  out[idx] = gelu_f(acc);
}

__global__ void copyq_kernel(const float* __restrict__ q, float* __restrict__ xb){
  long idx = (long)blockIdx.x*blockDim.x+threadIdx.x;
  long total=(long)BB*NQL*DD; if(idx>=total) return;
  long t = idx>>8; int c = (int)(idx&255);
  xb[t*D2 + DD + c] = q[idx];
}

__global__ __launch_bounds__(256) void rmsnorm_kernel(const float* __restrict__ x,
    const float* __restrict__ g, float* __restrict__ y){
  int wave=threadIdx.x>>5, lane=threadIdx.x&31;
  long rid = (long)blockIdx.x*8 + wave;
  const float* xr = x + rid*D2;
  float s=0.f;
  for(int j=lane;j<D2;j+=32){ float v=xr[j]; s+=v*v; }
  s = wave_sum(s);
  float nrm = fmaxf(sqrtf(s), 1e-12f);
  float sc = 22.627416997969522f / nrm;   // sqrt(512)
  float* yr = y + rid*D2;
  for(int j=lane;j<D2;j+=32) yr[j] = xr[j]*sc*g[j];
}

__global__ void gating_kernel(const float* __restrict__ xr, const float* __restrict__ gw,
                              int* __restrict__ tE, float* __restrict__ tGV){
  int t = blockIdx.x*blockDim.x + threadIdx.x;
  if (t >= BB*NQL) return;
  const float* x = xr + (long)t*D2;
  float lg[NEXP];
  #pragma unroll
  for(int e=0;e<NEXP;e++) lg[e]=0.f;
  for(int d=0;d<D2;d++){
    float xv = x[d];
    const float* row = gw + d*NEXP;
    #pragma unroll
    for(int e=0;e<NEXP;e++) lg[e] += xv*row[e];
  }
  float mx=lg[0];
  for(int e=1;e<NEXP;e++) mx=fmaxf(mx,lg[e]);
  float sum=0.f;
  for(int e=0;e<NEXP;e++){ lg[e]=expf(lg[e]-mx); sum+=lg[e]; }
  float inv=1.f/sum;
  for(int e=0;e<NEXP;e++) lg[e]*=inv;
  int i0=0; float g0=lg[0];
  for(int e=1;e<NEXP;e++) if(lg[e]>g0){ g0=lg[e]; i0=e; }
  int i1=-1; float g1=-1.f;
  for(int e=0;e<NEXP;e++) if(e!=i0 && lg[e]>g1){ g1=lg[e]; i1=e; }
  tE[2*t]=i0;  tGV[2*t]=g0;
  tE[2*t+1] = (g1 > 0.2f) ? i1 : -1;   // threshold on second choice
  tGV[2*t+1]=g1;
}

__global__ void routing_kernel(const int* __restrict__ tE, const float* __restrict__ tGV,
                               int* __restrict__ tSlot, float* __restrict__ tW){
  int b = threadIdx.x;
  if (b >= BB) return;
  int cnt[NEXP], kept[NEXP], cnt1[NEXP];
  for(int e=0;e<NEXP;e++){ cnt[e]=0; kept[e]=0; cnt1[e]=0; }
  for(int n=0;n<NQL;n++){
    int t=b*NQL+n; int e=tE[2*t];
    int pos = cnt[e]++;                 // cumsum over all first-choice routings
    int s = (pos<CAPC)?pos:-1;
    tSlot[2*t]=s; if(s>=0) kept[e]++;
  }
  for(int n=0;n<NQL;n++){
    int t=b*NQL+n; int e=tE[2*t+1]; int s=-1;
    if(e>=0){ int pos = cnt1[e]++ + kept[e]; if(pos<CAPC) s=pos; }
    tSlot[2*t+1]=s;
  }
  for(int n=0;n<NQL;n++){
    int t=b*NQL+n;
    float g0 = (tSlot[2*t]  >=0)? tGV[2*t]   : 0.f;
    float g1 = (tSlot[2*t+1]>=0)? tGV[2*t+1] : 0.f;
    float dn = g0+g1+1e-9f;
    tW[2*t]=g0/dn; tW[2*t+1]=g1/dn;
  }
}

__global__ __launch_bounds__(256) void scatter_kernel(const float* __restrict__ xr,
    const int* __restrict__ tE, const int* __restrict__ tSlot, float* __restrict__ xin){
  int wave=threadIdx.x>>5, lane=threadIdx.x&31;
  int id = blockIdx.x*8 + wave;         // 0..2*BB*NQL-1
  int t = id>>1, k = id&1;
  int s = tSlot[2*t+k];
  if (s<0) return;
  int e = tE[2*t+k];
  int b = t / NQL;
  const float4* src = (const float4*)(xr + (long)t*D2);
  float4* dst = (float4*)(xin + (((long)e*BB + b)*CAPC + s)*D2);
  #pragma unroll
  for(int j=0;j<4;j++) dst[lane + 32*j] = src[lane + 32*j];
}

__global__ __launch_bounds__(256) void expln_kernel(const float* __restrict__ xin,
   const float* __restrict__ lg, const float* __restrict__ lb, float* __restrict__ xln){
  int wave=threadIdx.x>>5, lane=threadIdx.x&31;
  long rid=(long)blockIdx.x*8+wave;      // NEXP*BB*CAPC rows
  int e = (int)(rid / (BB*CAPC));
  const float* xi = xin + rid*D2;
  float s=0.f;
  for(int j=lane;j<D2;j+=32) s+=xi[j];
  s=wave_sum(s); float mu=s*(1.f/D2);
  float v=0.f;
  for(int j=lane;j<D2;j+=32){ float d=xi[j]-mu; v+=d*d; }
  v=wave_sum(v)*(1.f/D2);
  float rs=rsqrtf(v+1e-5f);
  const float* ge=lg+(long)e*D2; const float* be=lb+(long)e*D2;
  float* yo=xln+rid*D2;
  for(int j=lane;j<D2;j+=32) yo[j]=(xi[j]-mu)*rs*ge[j]+be[j];
}

__global__ void combine_kernel(const float* __restrict__ xb, const float* __restrict__ eout,
   const int* __restrict__ tE, const int* __restrict__ tSlot, const float* __restrict__ tW,
   float* __restrict__ y){
  long idx=(long)blockIdx.x*blockDim.x+threadIdx.x;
  long total=(long)BB*NQL*128; if(idx>=total) return;
  long t=idx>>7; int j=(int)(idx&127);
  int b=(int)(t/NQL);
  float4 a=((const float4*)xb)[t*128+j];
  #pragma unroll
  for(int k=0;k<2;k++){
    int s=tSlot[2*t+k];
    if(s>=0){
      int e=tE[2*t+k]; float w=tW[2*t+k];
      float4 ev=((const float4*)eout)[(((long)e*BB+b)*CAPC+s)*128+j];
      a.x+=w*ev.x; a.y+=w*ev.y; a.z+=w*ev.z; a.w+=w*ev.w;
    }
  }
  ((float4*)y)[t*128+j]=a;
}

__global__ void add_kernel(const float* __restrict__ a, const float* __restrict__ b,
                           float* __restrict__ c, long n){
  long i=(long)blockIdx.x*blockDim.x+threadIdx.x; if(i<n) c[i]=a[i]+b[i];
}

__global__ __launch_bounds__(256) void out_kernel(const float* __restrict__ fus,
   const float* __restrict__ ow, const float* __restrict__ ob, float* __restrict__ out){
  int wave=threadIdx.x>>5, lane=threadIdx.x&31;
  long rid=(long)blockIdx.x*8+wave;
  const float* f=fus+rid*DD;
  float s=0.f;
  for(int j=lane;j<DD;j+=32) s+=f[j]*ow[j];
  s=wave_sum(s);
  if(lane==0) out[rid]=s+ob[0];
}

// ---------------- host side -------------------
struct BlockP {  // fields in alphabetical (pytree-flatten) order
  const float *b1,*b2,*bk,*bo,*bq,*bv,*gate_w,*ln_b,*ln_g,*lin_b,*lin_w,*rms_g,
              *topo_b,*topo_w,*val_b,*val_w,*w1,*w2,*wk,*wo,*wq,*wv;
};
static BlockP load_block(void* const* d_in, int& p){
  BlockP q;
  const float** f = (const float**)&q;
  for (int i=0;i<22;i++) f[i] = (const float*)d_in[p++];
  return q;
}

extern "C" void kernel_launch(void* const* d_in, const int* in_sizes, int n_in,
                              void* d_out, int out_size, void* d_ws, size_t ws_size,
                              hipStream_t stream) {
  (void)in_sizes; (void)n_in; (void)out_size; (void)ws_size;
  // pytree-flatten (alphabetical) order: W, X, X_mask, Z, params{fus_b,fus_w,gw{...},out_b,out_w,wx{...}}
  const float* W0 = (const float*)d_in[0];            // (1,B,64,48,48) -> W[0]
  const float* X  = (const float*)d_in[1];            // (B,1024,4)
  const unsigned char* Xmask = (const unsigned char*)d_in[2]; // (B,1024) bool
  const float* Z  = (const float*)d_in[3];            // (B,2048,3)
  int p = 4;
  const float* fus_b = (const float*)d_in[p++];
  const float* fus_w = (const float*)d_in[p++];
  BlockP gw = load_block(d_in, p);
  const float* out_b = (const float*)d_in[p++];
  const float* out_w = (const float*)d_in[p++];
  BlockP wx = load_block(d_in, p);

  const int NkMax = 2304;
  size_t off = 0;
  char* wsb = (char*)d_ws;
  auto alloc = [&](long n)->float*{
    float* r = (float*)(wsb + off);
    off = (off + (size_t)n*4 + 255) & ~(size_t)255;
    return r;
  };
  float* KEYS = alloc((long)BB*NkMax*DD);
  float* QUER = alloc((long)BB*NQL*DD);
  float* VALS = alloc((long)BB*NkMax*DD);
  float* QH   = alloc((long)BB*NQL*DD);
  float* KH   = alloc((long)BB*NkMax*DD);
  float* VH   = alloc((long)BB*NkMax*DD);
  float* SC   = alloc((long)BB*8*QCC*NkMax);
  float* ATT  = alloc((long)BB*NQL*DD);
  float* XB   = alloc((long)BB*NQL*D2);
  float* XR   = alloc((long)BB*NQL*D2);
  int*   tE   = (int*)alloc(2L*BB*NQL);
  float* tGV  = alloc(2L*BB*NQL);
  int*   tSlot= (int*)alloc(2L*BB*NQL);
  float* tW   = alloc(2L*BB*NQL);
  float* XIN  = alloc((long)NEXP*BB*CAPC*D2);
  float* XLN  = alloc((long)NEXP*BB*CAPC*D2);
  float* HB   = alloc((long)NEXP*BB*CAPC*HPAD);
  float* EOUT = alloc((long)NEXP*BB*CAPC*D2);
  float* OGW  = alloc((long)BB*NQL*DD);
  float* OWX  = alloc((long)BB*NQL*DD);

  auto run_block = [&](const BlockP& P, int Nk,
                       const float* keySrc, int keyMode, int keyStride,
                       const float* valSrc, int valMode, int valK,
                       const unsigned char* mask, float* outBlk){
    long MQ = (long)BB*NQL, MK = (long)BB*Nk;
    topo_kernel<<<dim3((unsigned)((MK*DD+255)/256)),256,0,stream>>>(keySrc,keyMode,Nk,keyStride,P.topo_w,P.topo_b,KEYS);
    topo_kernel<<<dim3((unsigned)((MQ*DD+255)/256)),256,0,stream>>>(Z,0,NQL,3,P.topo_w,P.topo_b,QUER);
    values_kernel<<<dim3((unsigned)((MK*DD+255)/256)),256,0,stream>>>(valSrc,valMode,Nk,valK,P.val_w,P.val_b,VALS);
    // q/k/v projections (WMMA, 32x32 per wave)
    gemm_wmma<0><<<dim3(DD/32,(unsigned)((MQ+255)/256),1),256,0,stream>>>(QUER,DD,0,P.wq,DD,0,P.bq,0,QH,DD,0,(int)MQ,DD,DD);
    gemm_wmma<0><<<dim3(DD/32,(unsigned)((MK+255)/256),1),256,0,stream>>>(KEYS,DD,0,P.wk,DD,0,P.bk,0,KH,DD,0,(int)MK,DD,DD);
    gemm_wmma<0><<<dim3(DD/32,(unsigned)((MK+255)/256),1),256,0,stream>>>(VALS,DD,0,P.wv,DD,0,P.bv,0,VH,DD,0,(int)MK,DD,DD);
    // attention, chunked over queries
    for(int q0=0; q0<NQL; q0+=QCC){
      attn_scores<<<dim3(Nk/32,1,BB*8),256,0,stream>>>(QH,KH,SC,NQL,Nk,q0);
      softmax_rows<<<dim3(BB*8*QCC/8),256,0,stream>>>(SC,Nk,mask);
      attn_av<<<dim3(1,1,BB*8),256,0,stream>>>(SC,VH,ATT,NQL,Nk,q0);
    }
    // output proj into x[:, :256], queries into x[:, 256:]
    gemm_wmma<0><<<dim3(DD/32,(unsigned)((MQ+255)/256),1),256,0,stream>>>(ATT,DD,0,P.wo,DD,0,P.bo,0,XB,D2,0,(int)MQ,DD,DD);
    copyq_kernel<<<dim3((unsigned)((MQ*DD+255)/256)),256,0,stream>>>(QUER,XB);
    rmsnorm_kernel<<<dim3((unsigned)(MQ/8)),256,0,stream>>>(XB,P.rms_g,XR);
    // MoE
    gating_kernel<<<dim3((unsigned)((MQ+255)/256)),256,0,stream>>>(XR,P.gate_w,tE,tGV);
    routing_kernel<<<1,BB,0,stream>>>(tE,tGV,tSlot,tW);
    hipMemsetAsync(XIN,0,(size_t)NEXP*BB*CAPC*D2*sizeof(float),stream);
    scatter_kernel<<<dim3((unsigned)(2*MQ/8)),256,0,stream>>>(XR,tE,tSlot,XIN);
    expln_kernel<<<dim3(NEXP*BB*CAPC/8),256,0,stream>>>(XIN,P.ln_g,P.ln_b,XLN);
    // expert FFN, batched over experts (WMMA)
    gemm_wmma<2><<<dim3((HID+31)/32,(BB*CAPC+255)/256,NEXP),256,0,stream>>>(
        XLN,D2,(long)BB*CAPC*D2, P.w1,HID,(long)D2*HID, P.b1,HID,
        HB,HPAD,(long)BB*CAPC*HPAD, BB*CAPC,HID,D2);
    gemm_wmma<0><<<dim3(D2/32,(BB*CAPC+255)/256,NEXP),256,0,stream>>>(
        HB,HPAD,(long)BB*CAPC*HPAD, P.w2,D2,(long)HID*D2, P.b2,D2,
        EOUT,D2,(long)BB*CAPC*D2, BB*CAPC,D2,HID);
    combine_kernel<<<dim3((unsigned)((MQ*128+255)/256)),256,0,stream>>>(XB,EOUT,tE,tSlot,tW,XR);
    // block output: gelu(x @ lin_w + lin_b)
    gemm_wmma<1><<<dim3(DD/32,(unsigned)((MQ+255)/256),1),256,0,stream>>>(XR,D2,0,P.lin_w,DD,0,P.lin_b,0,outBlk,DD,0,(int)MQ,DD,D2);
  };

  run_block(gw, 1024, X, 0, 4, X, 0, 4, Xmask, OGW);
  run_block(wx, 2304, W0, 2, 0, W0, 1, 64, nullptr, OWX);

  long MQ = (long)BB*NQL;
  add_kernel<<<dim3((unsigned)((MQ*DD+255)/256)),256,0,stream>>>(OGW,OWX,QUER,MQ*DD);
  gemm_wmma<1><<<dim3(DD/32,(unsigned)((MQ+255)/256),1),256,0,stream>>>(QUER,DD,0,fus_w,DD,0,fus_b,0,ATT,DD,0,(int)MQ,DD,DD);
  out_kernel<<<dim3((unsigned)(MQ/8)),256,0,stream>>>(ATT,out_w,out_b,(float*)d_out);
}